// PixelConv_34626026340634
// MI455X (gfx1250) — compile-verified
//
#include <hip/hip_runtime.h>

typedef float v2f __attribute__((ext_vector_type(2)));
typedef float v4f __attribute__((ext_vector_type(4)));
typedef float v8f __attribute__((ext_vector_type(8)));

#define CH    64
#define HH    128
#define WW    128
#define TAPS  9
#define S2    4
#define KCH   36                  // TAPS * S2 (kernel channel dim)
#define WTILE 32                  // w-pixels per block
#define FCOLS 35                  // 34 halo cols + 1 pad (odd stride -> conflict-free)
#define FROWS 3
#define FSTRIDE (FROWS * FCOLS)   // 105 floats per channel
#define LDS_FLOATS 8192           // 32 KB, reused as output staging

// Compile-time helpers on the flattened K index k in [0,36):
//   pixel p = k/9, tap t = k%9, dx = t/3, dy = t%3 (torch unfold order t = dx*3+dy)
#define P_OF(k)  ((k) / 9)
#define T_OF(k)  ((k) % 9)
#define AOFF(k)  ((T_OF(k) % 3) * FCOLS + P_OF(k) + (T_OF(k) / 3))  // + m*105 + wb
#define KOFF(k)  (T_OF(k) * 128 + P_OF(k))                          // + s*32  + wb

#if defined(__has_builtin)
#  if __has_builtin(__builtin_amdgcn_global_load_async_to_lds_b32) && \
      __has_builtin(__builtin_amdgcn_s_wait_asynccnt)
#    define USE_ASYNC_LDS 1
#  endif
#endif

typedef __attribute__((address_space(1))) int g_i32;   // global  (HIP: "__device__ int*")
typedef __attribute__((address_space(3))) int l_i32;   // LDS     (HIP: "__shared__ int*")

__device__ __forceinline__ void stage_b32(const float* __restrict__ g, float* l) {
#ifdef USE_ASYNC_LDS
    __builtin_amdgcn_global_load_async_to_lds_b32((g_i32*)g, (l_i32*)l, 0, 0);
#else
    *l = *g;
#endif
}

__global__ __launch_bounds__(256, 2)
void PixelConv_34626026340634_kernel(const float* __restrict__ feat,
                                     const float* __restrict__ kern,
                                     float* __restrict__ out) {
    __shared__ float lds[LDS_FLOATS];
    float* ldsf = lds;                    // 64 * 105 = 6720 floats
    float* ldsk = lds + CH * FSTRIDE;     // 36 * 32  = 1152 floats (7872 total)

    const int tid = threadIdx.x;
    const int blk = blockIdx.x;           // ((n*128)+h)*4 + wtile
    const int wt  = blk & 3;
    const int h   = (blk >> 2) & 127;
    const int n   = blk >> 9;
    const int w0  = wt * WTILE;

    // ---- Stage feature halo: rows h-1..h+1, cols w0-1..w0+32 (34 wide) ----
    for (int idx = tid; idx < CH * FROWS * 34; idx += 256) {
        int c   = idx / (FROWS * 34);
        int rem = idx % (FROWS * 34);
        int row = rem / 34;
        int col = rem % 34;
        int gh = h - 1 + row;
        int gw = w0 - 1 + col;
        float* dst = &ldsf[c * FSTRIDE + row * FCOLS + col];
        if (gh >= 0 && gh < HH && gw >= 0 && gw < WW)
            stage_b32(&feat[((n * CH + c) * HH + gh) * WW + gw], dst);
        else
            *dst = 0.f;
    }
    // ---- Stage per-pixel filters: 36 channels x 32 w (coalesced, in-bounds) ----
    for (int idx = tid; idx < KCH * WTILE; idx += 256) {
        int ch = idx >> 5;
        int w  = idx & 31;
        stage_b32(&kern[((n * KCH + ch) * HH + h) * WW + w0 + w], &ldsk[ch * WTILE + w]);
    }
#ifdef USE_ASYNC_LDS
    __builtin_amdgcn_s_wait_asynccnt(0);
#endif
    __syncthreads();

    const int  lane  = tid & 31;
    const int  wave  = tid >> 5;      // 8 waves, each owns 4 consecutive w pixels
    const int  m     = lane & 15;     // A: M row within tile; B/D: N column
    const bool hi    = lane >= 16;    // lane half selects K slots {0,1} vs {2,3}
    const int  pcol  = m >> 2;        // B column -> pixel within 4-group
    const int  s     = m & 3;         // B column -> sub-pixel index (s = 2r+q)
    const int  wb    = wave * 4;      // wave's pixel group base (local w)
    const int  abase = m * FSTRIDE + wb;
    const int  kbase = s * 32 + wb;

    v8f acc[4] = {v8f{}, v8f{}, v8f{}, v8f{}};

    #pragma unroll
    for (int j = 0; j < 9; ++j) {     // K = 36 via 9 chained K=4 WMMAs
        const int kL0 = 4 * j, kL1 = 4 * j + 1, kH0 = 4 * j + 2, kH1 = 4 * j + 3;
        // per-lane-half offsets: constants after unroll, one cndmask each
        const int a0  = hi ? AOFF(kH0) : AOFF(kL0);
        const int a1  = hi ? AOFF(kH1) : AOFF(kL1);
        const int kf0 = hi ? KOFF(kH0) : KOFF(kL0);
        const int kf1 = hi ? KOFF(kH1) : KOFF(kL1);
        const int bp0 = hi ? P_OF(kH0) : P_OF(kL0);
        const int bp1 = hi ? P_OF(kH1) : P_OF(kL1);

        // Block-diagonal B fragment: unconditional load, value select (no exec mask)
        const float bx = ldsk[kbase + kf0];
        const float by = ldsk[kbase + kf1];
        v2f b;
        b.x = (pcol == bp0) ? bx : 0.f;
        b.y = (pcol == bp1) ? by : 0.f;

        #pragma unroll
        for (int tile = 0; tile < 4; ++tile) {          // 4 x 16 channels
            v2f a;
            a.x = ldsf[tile * (16 * FSTRIDE) + abase + a0];
            a.y = ldsf[tile * (16 * FSTRIDE) + abase + a1];
            acc[tile] = __builtin_amdgcn_wmma_f32_16x16x4_f32(
                false, a, false, b, (short)0, acc[tile], false, false);
        }
    }

    // ---- Transpose through LDS for coalesced pixel-shuffled stores ----
    __syncthreads();                  // done reading ldsf/ldsk; reuse as staging
    {
        const int r = s >> 1;                     // s = 2r + q
        const int q = s & 1;
        const int colOut = 2 * (wb + pcol) + q;   // 0..63 within block's out rows
        #pragma unroll
        for (int tile = 0; tile < 4; ++tile) {
            #pragma unroll
            for (int v = 0; v < 8; ++v) {
                int c = tile * 16 + v + (hi ? 8 : 0);
                lds[c * 128 + r * 64 + colOut] = acc[tile][v];
            }
        }
    }
    __syncthreads();

    // stage[c][r][col64] -> out[n, c, 2h+r, 2*w0 + col]; 16B vectors, NT stores
    for (int idx = tid; idx < LDS_FLOATS / 4; idx += 256) {
        int c  = idx >> 5;            // 32 float4 per channel (2 rows x 16)
        int r  = (idx >> 4) & 1;
        int c4 = (idx & 15) * 4;
        v4f v = *(const v4f*)&lds[idx * 4];
        v4f* dst = (v4f*)&out[((n * CH + c) * (2 * HH) + (2 * h + r)) * (2 * WW)
                              + (2 * w0 + c4)];
        __builtin_nontemporal_store(v, dst);
    }
}

extern "C" void kernel_launch(void* const* d_in, const int* in_sizes, int n_in,
                              void* d_out, int out_size, void* d_ws, size_t ws_size,
                              hipStream_t stream) {
    (void)in_sizes; (void)n_in; (void)d_ws; (void)ws_size; (void)out_size;
    const float* feat = (const float*)d_in[0];   // (8, 64, 128, 128) f32
    const float* kern = (const float*)d_in[1];   // (8, 36, 128, 128) f32
    float* out = (float*)d_out;                  // (8, 64, 256, 256) f32

    dim3 grid(8 * 128 * 4);   // n * h * (W/32)
    dim3 block(256);          // 8 wave32
    PixelConv_34626026340634_kernel<<<grid, block, 0, stream>>>(feat, kern, out);
}